// AAMLoss_75436805587459
// MI455X (gfx1250) — compile-verified
//
#include <hip/hip_runtime.h>
#include <cstdint>

// AAM / CosFace loss: N=8192 rows, C=32000 cols, fp32.
// MI455X plan:
//   - 1 workgroup (256 thr = 8 wave32) per row; dynamic LDS = 128 KB row buffer
//     (2 blocks co-resident per 320 KB WGP).
//   - Whole row streamed global->LDS via global_load_async_to_lds_b128 using one
//     base address pair + immediate offsets (IOFFSET applies to both sides).
//   - Pass 1 pipelined against the async fill: chunk k consumed after
//     s_wait_asynccnt (30-k)  (async loads complete in order per wave; each lane
//     reads only the bytes it issued itself).
//   - Pass 2 (sum exp) from LDS; per-row NLL -> workspace; deterministic
//     tree-sum kernel -> mean. HBM traffic = 1.02 GB once ~= 44 us floor.

#define N_COLS   32000
#define NV4      (N_COLS / 4)          // 8000 float4 per row
#define BLOCK    256
#define NWAVES   (BLOCK / 32)          // 8 waves (wave32)
#define NFULL    (NV4 / BLOCK)         // 31 full chunks of 256 float4
#define NTAIL    (NV4 - NFULL * BLOCK) // 64 remaining float4
#define CHUNK_B  (BLOCK * 16)          // 4096 bytes per chunk
#define MARGIN_S 15.0f                 // SCALE * MARGIN = 30 * 0.5
#define SCALE_F  30.0f
#define EPS_F    1e-12f

__device__ __forceinline__ float wave_sum(float v) {
#pragma unroll
    for (int m = 16; m >= 1; m >>= 1) v += __shfl_xor(v, m, 32);
    return v;
}
__device__ __forceinline__ float wave_max(float v) {
#pragma unroll
    for (int m = 16; m >= 1; m >>= 1) v = fmaxf(v, __shfl_xor(v, m, 32));
    return v;
}

extern "C" __global__ __launch_bounds__(BLOCK, 2)
void aam_row_kernel(const float* __restrict__ logits,
                    const int*   __restrict__ labels,
                    float*       __restrict__ row_nll) {
    extern __shared__ __align__(16) float lds[];   // N_COLS data + 32 scratch floats
    float* scratch = lds + N_COLS;

    const int row  = blockIdx.x;
    const int tid  = threadIdx.x;
    const int wave = tid >> 5;
    const int lane = tid & 31;
    const float* __restrict__ src = logits + (size_t)row * N_COLS;

    // ---- Issue the whole row as async global->LDS copies (ASYNCcnt) ----------
    // One base pair per thread; instruction IOFFSET advances BOTH addresses.
    const float* gbase = src + tid * 4;                       // tid*16 bytes
    unsigned     lbase = (unsigned)(uintptr_t)(lds + tid * 4);// low 32b = LDS off
#pragma unroll
    for (int k = 0; k < NFULL; ++k) {
        asm volatile("global_load_async_to_lds_b128 %0, %1, off offset:%2"
                     :: "v"(lbase), "v"(gbase), "i"(k * CHUNK_B) : "memory");
    }
    if (tid < NTAIL) {
        asm volatile("global_load_async_to_lds_b128 %0, %1, off offset:%2"
                     :: "v"(lbase), "v"(gbase), "i"(NFULL * CHUNK_B) : "memory");
    }

    // ---- Pass 1, pipelined: consume chunk k once ASYNCcnt <= NFULL-1-k -------
    // (per-wave in-order completion; each lane reads only its own 16B slots;
    //  waves 2..7 issued NFULL, waves 0..1 issued NFULL+1 -> NFULL-1-k is safe)
    float sumsq = 0.0f;
    float mx    = -INFINITY;
#pragma unroll
    for (int k = 0; k < NFULL; ++k) {
        asm volatile("s_wait_asynccnt %0" :: "i"(NFULL - 1 - k) : "memory");
        float4 d = ((const float4*)lds)[tid + k * BLOCK];     // ds_load_b128
        sumsq += d.x * d.x + d.y * d.y + d.z * d.z + d.w * d.w;
        mx = fmaxf(mx, fmaxf(fmaxf(d.x, d.y), fmaxf(d.z, d.w)));
    }
    asm volatile("s_wait_asynccnt 0" ::: "memory");           // tail chunk too
    if (tid < NTAIL) {
        float4 d = ((const float4*)lds)[tid + NFULL * BLOCK];
        sumsq += d.x * d.x + d.y * d.y + d.z * d.z + d.w * d.w;
        mx = fmaxf(mx, fmaxf(fmaxf(d.x, d.y), fmaxf(d.z, d.w)));
    }

    sumsq = wave_sum(sumsq);
    mx    = wave_max(mx);
    if (lane == 0) { scratch[wave] = sumsq; scratch[NWAVES + wave] = mx; }
    __syncthreads();   // also publishes all waves' LDS row data block-wide

    if (tid == 0) {
        float s = 0.0f, m = -INFINITY;
#pragma unroll
        for (int w = 0; w < NWAVES; ++w) {
            s += scratch[w];
            m  = fmaxf(m, scratch[NWAVES + w]);
        }
        const int   lab = labels[row];
        const float xl  = lds[lab];
        const float a   = SCALE_F / fmaxf(sqrtf(s), EPS_F);   // a > 0
        // margin only lowers the label logit => scaled row max = a*m
        scratch[16] = a;
        scratch[17] = a * m;                 // M
        scratch[18] = a * xl - MARGIN_S;     // scaled label logit w/ margin
        scratch[19] = a * xl;                // scaled label logit w/o margin
    }
    __syncthreads();
    const float a    = scratch[16];
    const float M    = scratch[17];
    const float zl_m = scratch[18];
    const float zl   = scratch[19];

    // ---- Pass 2 (LDS-resident): sum of exp(a*x - M) --------------------------
    float se = 0.0f;
    for (int v = tid; v < NV4; v += BLOCK) {
        float4 d = ((const float4*)lds)[v];
        se += __expf(fmaf(a, d.x, -M)) + __expf(fmaf(a, d.y, -M))
            + __expf(fmaf(a, d.z, -M)) + __expf(fmaf(a, d.w, -M));
    }
    se = wave_sum(se);
    if (lane == 0) scratch[20 + wave] = se;
    __syncthreads();

    if (tid == 0) {
        float s = 0.0f;
#pragma unroll
        for (int w = 0; w < NWAVES; ++w) s += scratch[20 + w];
        // swap the label term for its margin-adjusted version
        s += __expf(zl_m - M) - __expf(zl - M);
        row_nll[row] = M + __logf(s) - zl_m;   // = logsumexp - label_logit
    }
}

extern "C" __global__ __launch_bounds__(BLOCK)
void mean_kernel(const float* __restrict__ row_nll, float* __restrict__ out, int n) {
    __shared__ float sc[NWAVES];
    float s = 0.0f;
    for (int i = threadIdx.x; i < n; i += BLOCK) s += row_nll[i];
    s = wave_sum(s);
    if ((threadIdx.x & 31) == 0) sc[threadIdx.x >> 5] = s;
    __syncthreads();
    if (threadIdx.x == 0) {
        float t = 0.0f;
#pragma unroll
        for (int w = 0; w < NWAVES; ++w) t += sc[w];
        out[0] = t / (float)n;
    }
}

extern "C" void kernel_launch(void* const* d_in, const int* in_sizes, int n_in,
                              void* d_out, int out_size, void* d_ws, size_t ws_size,
                              hipStream_t stream) {
    const float* logits = (const float*)d_in[0];
    const int*   labels = (const int*)d_in[1];
    float*       out    = (float*)d_out;
    float*       rows   = (float*)d_ws;          // n floats of scratch

    const int n = in_sizes[1];                   // 8192 rows (C fixed at 32000)
    const size_t lds_bytes = (size_t)(N_COLS + 32) * sizeof(float);  // 128128 B

    aam_row_kernel<<<dim3(n), dim3(BLOCK), lds_bytes, stream>>>(logits, labels, rows);
    mean_kernel<<<dim3(1), dim3(BLOCK), 0, stream>>>(rows, out, n);
}